// ModelGATConv_44126493999616
// MI455X (gfx1250) — compile-verified
//
#include <hip/hip_runtime.h>
#include <cstddef>
#include <cstdint>

typedef __attribute__((ext_vector_type(2))) float v2f;
typedef __attribute__((ext_vector_type(8))) float v8f;

// ---------------- utility fills ----------------
__global__ __launch_bounds__(256) void fill_f32(float* p, float v, long long n) {
  long long i = (long long)blockIdx.x * blockDim.x + threadIdx.x;
  if (i < n) p[i] = v;
}
__global__ __launch_bounds__(256) void fill_u32(unsigned* p, unsigned v, long long n) {
  long long i = (long long)blockIdx.x * blockDim.x + threadIdx.x;
  if (i < n) p[i] = v;
}

// order-preserving float<->uint for atomicMax-based segment max
__device__ __forceinline__ unsigned f32_ord(float f) {
  unsigned u = __float_as_uint(f);
  return (u & 0x80000000u) ? ~u : (u | 0x80000000u);
}
__device__ __forceinline__ float ord_f32(unsigned u) {
  return (u & 0x80000000u) ? __uint_as_float(u & 0x7FFFFFFFu) : __uint_as_float(~u);
}

// ---------------- layer 0: 1 -> 64 (rank-1) ----------------
__global__ __launch_bounds__(256) void layer0_proj(
    const float* __restrict__ x, const float* __restrict__ W0,
    const float* __restrict__ a_s, const float* __restrict__ a_d,
    float* __restrict__ h, float* __restrict__ ss, float* __restrict__ sd, int n) {
  int node = blockIdx.x * 256 + threadIdx.x;
  if (node >= n) return;
  float xv = x[node];
  float accs = 0.f, accd = 0.f;
  #pragma unroll 8
  for (int j = 0; j < 64; ++j) {
    float hv = xv * W0[j];
    h[(size_t)node * 64 + j] = hv;
    accs += hv * a_s[j];
    accd += hv * a_d[j];
  }
  ss[node] = accs;
  sd[node] = accd;
}

// ---------------- WMMA f32 GEMM: h[N,64] = act[N,64] @ W[64,64] ----------------
// One wave computes a 16-row x 64-col tile with 64x v_wmma_f32_16x16x4_f32.
// - Out-of-range rows are address-clamped (not branched): results garbage but
//   never stored -> fully convergent inner loop.
// - W is staged in LDS pre-paired: Wp[kpair][j] = {W[2kp][j], W[2kp+1][j]} so a
//   single aligned ds_load_b64 yields a WMMA-ready SRC1 VGPR pair (no repacking
//   movs). kpair stride padded to 160 floats so the two half-waves (adjacent
//   kpairs) land on disjoint 32-bank halves -> conflict-free b64 LDS reads.
#define KP_STRIDE 160
__global__ __launch_bounds__(256) void gemm64_wmma(
    const float* __restrict__ act, const float* __restrict__ W,
    float* __restrict__ h, int n_nodes) {
  __shared__ float Wsh[32 * KP_STRIDE];  // 20 KB
  for (int i = threadIdx.x; i < 64 * 64; i += 256) {
    int k = i >> 6, j = i & 63;
    Wsh[(k >> 1) * KP_STRIDE + j * 2 + (k & 1)] = W[i];
  }
  __syncthreads();

  const int lane = threadIdx.x & 31;
  const int wave = threadIdx.x >> 5;
  const int rowBase = blockIdx.x * 128 + wave * 16;
  const int lrow = lane & 15;
  const int hi = lane >> 4;          // 0: lanes 0-15, 1: lanes 16-31
  int r = rowBase + lrow;
  if (r >= n_nodes) r = n_nodes - 1; // clamp: garbage rows computed but never stored
  const float* __restrict__ arow = act + (size_t)r * 64 + hi * 2;
  // B base: half-wave hi reads kpair (k0/2 + hi); lane's column pair at j*2
  const float* __restrict__ bbase = Wsh + hi * KP_STRIDE + lrow * 2;

  v8f acc0 = {}, acc1 = {}, acc2 = {}, acc3 = {};

  #pragma unroll
  for (int k0 = 0; k0 < 64; k0 += 4) {
    // A fragment (16x4 f32): lanes 0-15 hold K=k0,k0+1; lanes 16-31 hold K=k0+2,k0+3
    v2f a = *(const v2f*)(arow + k0);                       // global_load_b64
    const float* bp = bbase + (k0 >> 1) * KP_STRIDE;
    v2f b0 = *(const v2f*)(bp + 0 * 32);                    // ds_load_b64 x4
    v2f b1 = *(const v2f*)(bp + 1 * 32);
    v2f b2 = *(const v2f*)(bp + 2 * 32);
    v2f b3 = *(const v2f*)(bp + 3 * 32);
    acc0 = __builtin_amdgcn_wmma_f32_16x16x4_f32(false, a, false, b0, (short)0, acc0, false, false);
    acc1 = __builtin_amdgcn_wmma_f32_16x16x4_f32(false, a, false, b1, (short)0, acc1, false, false);
    acc2 = __builtin_amdgcn_wmma_f32_16x16x4_f32(false, a, false, b2, (short)0, acc2, false, false);
    acc3 = __builtin_amdgcn_wmma_f32_16x16x4_f32(false, a, false, b3, (short)0, acc3, false, false);
  }

  // C/D layout: VGPR i, lanes 0-15: row M=i, col=lane; lanes 16-31: row M=8+i, col=lane-16
  const int rowOff = rowBase + (hi ? 8 : 0);
  #pragma unroll
  for (int i = 0; i < 8; ++i) {
    int row = rowOff + i;
    if (row < n_nodes) {
      float* hp = h + (size_t)row * 64 + lrow;
      hp[0]  = acc0[i];
      hp[16] = acc1[i];
      hp[32] = acc2[i];
      hp[48] = acc3[i];
    }
  }
}

// ---------------- per-node attention scores: ss = h.a_s, sd = h.a_d ----------------
__global__ __launch_bounds__(256) void node_scores(
    const float* __restrict__ h, const float* __restrict__ a_s, const float* __restrict__ a_d,
    float* __restrict__ ss, float* __restrict__ sd, int n) {
  int node = blockIdx.x * 8 + (threadIdx.x >> 5);
  if (node >= n) return;
  int lane = threadIdx.x & 31;
  float h0 = h[(size_t)node * 64 + lane];
  float h1 = h[(size_t)node * 64 + lane + 32];
  float ps = h0 * a_s[lane] + h1 * a_s[lane + 32];
  float pd = h0 * a_d[lane] + h1 * a_d[lane + 32];
  #pragma unroll
  for (int off = 16; off; off >>= 1) {
    ps += __shfl_xor(ps, off, 32);
    pd += __shfl_xor(pd, off, 32);
  }
  if (lane == 0) { ss[node] = ps; sd[node] = pd; }
}

// ---------------- edge pass 1: logits + segment max ----------------
__global__ __launch_bounds__(256) void edge_logits_max(
    const int* __restrict__ esrc, const int* __restrict__ edst,
    const float* __restrict__ ss, const float* __restrict__ sd,
    float* __restrict__ ebuf, unsigned* __restrict__ menc,
    long long E, long long total) {
  long long i = (long long)blockIdx.x * blockDim.x + threadIdx.x;
  if (i >= total) return;
  int s, d;
  if (i < E) { s = esrc[i]; d = edst[i]; } else { s = d = (int)(i - E); }
  float e = ss[s] + sd[d];
  e = (e > 0.f) ? e : 0.2f * e;   // LeakyReLU(0.2)
  ebuf[i] = e;
  atomicMax(menc + d, f32_ord(e));
}

// ---------------- edge pass 2: exp + segment sum ----------------
__global__ __launch_bounds__(256) void edge_exp_sum(
    const int* __restrict__ edst, float* __restrict__ ebuf,
    const unsigned* __restrict__ menc, float* __restrict__ denom,
    long long E, long long total) {
  long long i = (long long)blockIdx.x * blockDim.x + threadIdx.x;
  if (i >= total) return;
  int d = (i < E) ? edst[i] : (int)(i - E);
  float m = ord_f32(menc[d]);
  float ex = __expf(ebuf[i] - m);
  ebuf[i] = ex;
  atomicAdd(denom + d, ex);
}

// ---------------- edge pass 3 (64-dim): out[dst] += alpha * h[src] ----------------
// One wave per edge: 2 components per lane -> one coalesced 128B row per access.
__global__ __launch_bounds__(256) void edge_aggregate64(
    const int* __restrict__ esrc, const int* __restrict__ edst,
    const float* __restrict__ ebuf, const float* __restrict__ denom,
    const float* __restrict__ h, float* __restrict__ out,
    long long E, long long total) {
  long long e = (long long)blockIdx.x * 8 + (threadIdx.x >> 5);
  if (e >= total) return;
  int lane = threadIdx.x & 31;
  int s, d;
  if (e < E) { s = esrc[e]; d = edst[e]; } else { s = d = (int)(e - E); }
  float alpha = ebuf[e] / denom[d];
  float h0 = h[(size_t)s * 64 + lane];
  float h1 = h[(size_t)s * 64 + lane + 32];
  atomicAdd(out + (size_t)d * 64 + lane, alpha * h0);
  atomicAdd(out + (size_t)d * 64 + lane + 32, alpha * h1);
}

// ---------------- edge pass 3 (1-dim, last layer) ----------------
__global__ __launch_bounds__(256) void edge_aggregate1(
    const int* __restrict__ esrc, const int* __restrict__ edst,
    const float* __restrict__ ebuf, const float* __restrict__ denom,
    const float* __restrict__ hL, float* __restrict__ outL,
    long long E, long long total) {
  long long i = (long long)blockIdx.x * blockDim.x + threadIdx.x;
  if (i >= total) return;
  int s, d;
  if (i < E) { s = esrc[i]; d = edst[i]; } else { s = d = (int)(i - E); }
  float alpha = ebuf[i] / denom[d];
  atomicAdd(outL + d, alpha * hL[s]);
}

// ---------------- epilogue: bias + relu, in place ----------------
__global__ __launch_bounds__(256) void epilogue_relu(
    float* __restrict__ out, const float* __restrict__ b, long long n64) {
  long long i = (long long)blockIdx.x * blockDim.x + threadIdx.x;
  if (i >= n64) return;
  float v = out[i] + b[i & 63];
  out[i] = (v > 0.f) ? v : 0.f;
}

// ---------------- last layer projection 64 -> 1 (wave per node) ----------------
__global__ __launch_bounds__(256) void last_proj(
    const float* __restrict__ act, const float* __restrict__ WL,
    const float* __restrict__ asL, const float* __restrict__ adL,
    float* __restrict__ hL, float* __restrict__ ss, float* __restrict__ sd, int n) {
  int node = blockIdx.x * 8 + (threadIdx.x >> 5);
  if (node >= n) return;
  int lane = threadIdx.x & 31;
  float p = act[(size_t)node * 64 + lane] * WL[lane] +
            act[(size_t)node * 64 + lane + 32] * WL[lane + 32];
  #pragma unroll
  for (int off = 16; off; off >>= 1) p += __shfl_xor(p, off, 32);
  if (lane == 0) {
    hL[node] = p;
    ss[node] = p * asL[0];
    sd[node] = p * adL[0];
  }
}

// ---------------- final epilogue: bias + sigmoid ----------------
__global__ __launch_bounds__(256) void last_epilogue(
    const float* __restrict__ outL, const float* __restrict__ bL,
    float* __restrict__ out, int n) {
  int i = blockIdx.x * 256 + threadIdx.x;
  if (i >= n) return;
  float v = outL[i] + bL[0];
  out[i] = 1.f / (1.f + __expf(-v));
}

// ================= host orchestration =================
extern "C" void kernel_launch(void* const* d_in, const int* in_sizes, int n_in,
                              void* d_out, int out_size, void* d_ws, size_t ws_size,
                              hipStream_t stream) {
  const float* x    = (const float*)d_in[0];
  const int*   eidx = (const int*)d_in[1];
  // d_in[2] = edge_weight: ignored (edge_dim=None in reference)
  const float* W0   = (const float*)d_in[3];
  const float* as0  = (const float*)d_in[4];
  const float* ad0  = (const float*)d_in[5];
  const float* b0   = (const float*)d_in[6];
  const float* Wm   = (const float*)d_in[7];
  const float* asm_ = (const float*)d_in[8];
  const float* adm  = (const float*)d_in[9];
  const float* bm   = (const float*)d_in[10];
  const float* WL   = (const float*)d_in[11];
  const float* asL  = (const float*)d_in[12];
  const float* adL  = (const float*)d_in[13];
  const float* bL   = (const float*)d_in[14];
  float* out = (float*)d_out;

  const int       N = in_sizes[0];
  const long long E = (long long)in_sizes[1] / 2;
  const long long T = E + N;                 // edges incl. self loops
  const int* esrc = eidx;
  const int* edst = eidx + E;

  // workspace layout
  float* base = (float*)d_ws;
  const size_t N64 = (size_t)N * 64;
  float* B[3] = { base, base + N64, base + 2 * N64 };
  float* ebuf = base + 3 * N64;              // T floats
  float* ss   = ebuf + T;                    // N
  float* sd   = ss + N;                      // N
  unsigned* menc = (unsigned*)(sd + N);      // N
  float* denom = (float*)(menc + N);         // N
  float* hL    = denom + N;                  // N
  float* outL  = hL + N;                     // N

  dim3 blk(256);
  auto g = [](long long n, int per) { return dim3((unsigned)((n + per - 1) / per)); };

  // ---- layer 0: 1 -> 64 ----
  layer0_proj<<<g(N, 256), blk, 0, stream>>>(x, W0, as0, ad0, B[0], ss, sd, N);
  fill_u32<<<g(N, 256), blk, 0, stream>>>(menc, 0u, N);
  fill_f32<<<g(N, 256), blk, 0, stream>>>(denom, 0.f, N);
  edge_logits_max<<<g(T, 256), blk, 0, stream>>>(esrc, edst, ss, sd, ebuf, menc, E, T);
  edge_exp_sum<<<g(T, 256), blk, 0, stream>>>(edst, ebuf, menc, denom, E, T);
  fill_f32<<<g((long long)N64, 256), blk, 0, stream>>>(B[1], 0.f, (long long)N64);
  edge_aggregate64<<<g(T, 8), blk, 0, stream>>>(esrc, edst, ebuf, denom, B[0], B[1], E, T);
  epilogue_relu<<<g((long long)N64, 256), blk, 0, stream>>>(B[1], b0, (long long)N64);

  // ---- middle layers: 64 -> 64 (x3), WMMA GEMM ----
  int act = 1, hb = 0, ob = 2;
  for (int L = 0; L < 3; ++L) {
    gemm64_wmma<<<g(N, 128), blk, 0, stream>>>(B[act], Wm + (size_t)L * 4096, B[hb], N);
    node_scores<<<g(N, 8), blk, 0, stream>>>(B[hb], asm_ + L * 64, adm + L * 64, ss, sd, N);
    fill_u32<<<g(N, 256), blk, 0, stream>>>(menc, 0u, N);
    fill_f32<<<g(N, 256), blk, 0, stream>>>(denom, 0.f, N);
    edge_logits_max<<<g(T, 256), blk, 0, stream>>>(esrc, edst, ss, sd, ebuf, menc, E, T);
    edge_exp_sum<<<g(T, 256), blk, 0, stream>>>(edst, ebuf, menc, denom, E, T);
    fill_f32<<<g((long long)N64, 256), blk, 0, stream>>>(B[ob], 0.f, (long long)N64);
    edge_aggregate64<<<g(T, 8), blk, 0, stream>>>(esrc, edst, ebuf, denom, B[hb], B[ob], E, T);
    epilogue_relu<<<g((long long)N64, 256), blk, 0, stream>>>(B[ob], bm + L * 64, (long long)N64);
    int na = ob, nh = act, no = hb;
    act = na; hb = nh; ob = no;
  }

  // ---- last layer: 64 -> 1, sigmoid ----
  last_proj<<<g(N, 8), blk, 0, stream>>>(B[act], WL, asL, adL, hL, ss, sd, N);
  fill_u32<<<g(N, 256), blk, 0, stream>>>(menc, 0u, N);
  fill_f32<<<g(N, 256), blk, 0, stream>>>(denom, 0.f, N);
  edge_logits_max<<<g(T, 256), blk, 0, stream>>>(esrc, edst, ss, sd, ebuf, menc, E, T);
  edge_exp_sum<<<g(T, 256), blk, 0, stream>>>(edst, ebuf, menc, denom, E, T);
  fill_f32<<<g(N, 256), blk, 0, stream>>>(outL, 0.f, N);
  edge_aggregate1<<<g(T, 256), blk, 0, stream>>>(esrc, edst, ebuf, denom, hL, outL, E, T);
  last_epilogue<<<g(N, 256), blk, 0, stream>>>(outL, bL, out, N);
}